// CrossAttention_53541062312049
// MI455X (gfx1250) — compile-verified
//
#include <hip/hip_runtime.h>

typedef float v2f __attribute__((ext_vector_type(2)));
typedef float v8f __attribute__((ext_vector_type(8)));

#define CDIM 256
#define NTOK 4096
#define BSZ  2
#define LOG2E 1.4426950408889634f
#define TILEW 20            // padded row width for [c][n16] LDS tiles
#define NT16  (NTOK / 16)

#define LDS_FENCE()   __asm__ volatile("s_wait_dscnt 0" ::: "memory")
#define SCHED_FENCE() __asm__ volatile("" ::: "memory")
// previous tile's 16 async b128 done; next tile's 16 may still be in flight
#define ASYNC_WAIT_PREFETCHED() __asm__ volatile("s_wait_asynccnt 0x10" ::: "memory")
#define ASYNC_WAIT_ALL()        __asm__ volatile("s_wait_asynccnt 0x0"  ::: "memory")

__device__ __forceinline__ v8f wmma4(v2f a, v2f b, v8f c) {
  // D(16x16,f32) = A(16x4,f32) x B(4x16,f32) + C
  return __builtin_amdgcn_wmma_f32_16x16x4_f32(false, a, false, b, (short)0, c,
                                               false, false);
}

__device__ __forceinline__ v8f vzero8() {
  v8f z = {0.f, 0.f, 0.f, 0.f, 0.f, 0.f, 0.f, 0.f};
  return z;
}

// Async DMA: 16B global -> LDS, tracked by ASYNCcnt (no VGPR staging).
// Flat->LDS mapping truncates to low 32 bits, so low bits of the generic
// shared pointer are the LDS byte address.
__device__ __forceinline__ void async_copy_b128(float* lds_dst,
                                                const float* gsrc) {
  unsigned ldsoff = (unsigned)(unsigned long long)lds_dst;
  __asm__ volatile("global_load_async_to_lds_b128 %0, %1, off"
                   :: "v"(ldsoff), "v"(gsrc)
                   : "memory");
}

// ---------------------------------------------------------------------------
// Kernel 1: K = Wk*t + bk, V = Wv*t + bv   (per-token 1x1x1 conv == GEMM)
// Grid: (NTOK/64, B, 2), 128 threads (4 waves). Wave w owns o-range [w*64,+64).
// ---------------------------------------------------------------------------
__global__ __launch_bounds__(128) void kv_proj_kernel(
    const float* __restrict__ t,
    const float* __restrict__ wk, const float* __restrict__ bk,
    const float* __restrict__ wv, const float* __restrict__ bv,
    float* __restrict__ Kout, float* __restrict__ Vout)
{
  const int nb  = blockIdx.x;
  const int b   = blockIdx.y;
  const int mat = blockIdx.z;
  const float* W    = mat ? wv : wk;
  const float* bias = mat ? bv : bk;
  float* Out        = (mat ? Vout : Kout) + (size_t)b * CDIM * NTOK;
  const float* tb   = t + (size_t)b * CDIM * NTOK;

  const int lane = threadIdx.x & 31;
  const int wave = threadIdx.x >> 5;
  const int half = lane >> 4;
  const int l15  = lane & 15;
  const int n0    = nb * 64;
  const int obase = wave * 64;

  v8f acc[4][4];
#pragma unroll
  for (int i = 0; i < 4; ++i)
#pragma unroll
    for (int j = 0; j < 4; ++j) acc[i][j] = vzero8();

  for (int k = 0; k < CDIM; k += 4) {
    const int kk = k + 2 * half;   // this lane-half's K pair {kk, kk+1}
    v2f afr[4], bfr[4];
#pragma unroll
    for (int ot = 0; ot < 4; ++ot)  // A[o][k] = W[o][c], row o = l15
      afr[ot] = *(const v2f*)(W + (size_t)(obase + ot * 16 + l15) * CDIM + kk);
#pragma unroll
    for (int nt = 0; nt < 4; ++nt) { // B[k][n] = t[c][n], col n = l15
      const int n = n0 + nt * 16 + l15;
      bfr[nt][0] = tb[(size_t)(kk + 0) * NTOK + n];
      bfr[nt][1] = tb[(size_t)(kk + 1) * NTOK + n];
    }
#pragma unroll
    for (int ot = 0; ot < 4; ++ot)
#pragma unroll
      for (int nt = 0; nt < 4; ++nt)
        acc[ot][nt] = wmma4(afr[ot], bfr[nt], acc[ot][nt]);
  }

#pragma unroll
  for (int ot = 0; ot < 4; ++ot)
#pragma unroll
    for (int r = 0; r < 8; ++r) {
      const int o = obase + ot * 16 + r + 8 * half;  // D row
      const float bb = bias[o];
#pragma unroll
      for (int nt = 0; nt < 4; ++nt)
        Out[(size_t)o * NTOK + n0 + nt * 16 + l15] = acc[ot][nt][r] + bb;
    }
}

// ---------------------------------------------------------------------------
// Kernel 2: fused Q-projection + flash attention + output projection + residual
// Grid: (NTOK/64, B), 128 threads (4 waves). Wave w owns query rows m0+w*16..+15.
// Dynamic LDS: double-buffered K/V tiles [256][20] each, filled by async DMA.
// ---------------------------------------------------------------------------
__global__ __launch_bounds__(128, 1) void attn_kernel(
    const float* __restrict__ x,
    const float* __restrict__ wq, const float* __restrict__ bq,
    const float* __restrict__ wo, const float* __restrict__ bo,
    const float* __restrict__ Kin, const float* __restrict__ Vin,
    float* __restrict__ out)
{
  extern __shared__ float dynsmem[];
  float* Kbuf0 = dynsmem;                    // [256][TILEW]
  float* Vbuf0 = Kbuf0 + CDIM * TILEW;
  float* Kbuf1 = Vbuf0 + CDIM * TILEW;
  float* Vbuf1 = Kbuf1 + CDIM * TILEW;
  __shared__ float pbuf[4][16][18];          // per-wave fragment relayout

  const int mb = blockIdx.x;
  const int b  = blockIdx.y;
  const int lane = threadIdx.x & 31;
  const int wave = threadIdx.x >> 5;
  const int half = lane >> 4;
  const int l15  = lane & 15;
  const int m0 = mb * 64;
  const int mg = m0 + wave * 16 + l15;   // this lane's query token (B-frag col)

  const float* xb = x   + (size_t)b * CDIM * NTOK;
  const float* Kb = Kin + (size_t)b * CDIM * NTOK;
  const float* Vb = Vin + (size_t)b * CDIM * NTOK;

  const int n4   = (threadIdx.x & 3) * 4;    // staging: 4-float column group
  const int crow = threadIdx.x >> 2;         // staging: starting channel row

  // kick off async staging of key/value tile 0 while we compute Q
  {
#pragma unroll 8
    for (int cc = crow; cc < CDIM; cc += 32) {
      async_copy_b128(Kbuf0 + cc * TILEW + n4, Kb + (size_t)cc * NTOK + n4);
      async_copy_b128(Vbuf0 + cc * TILEW + n4, Vb + (size_t)cc * NTOK + n4);
    }
  }

  // ---- Phase A: Q projection for this wave's 16 tokens; keep Q^T as A-frags ----
  v2f a_q[64];                         // a_q[j] covers c = 4j..4j+3
#pragma unroll
  for (int og = 0; og < 4; ++og) {
    v8f qa[4];
#pragma unroll
    for (int i = 0; i < 4; ++i) qa[i] = vzero8();
    for (int k = 0; k < CDIM; k += 4) {
      const int kk = k + 2 * half;
      v2f bfr;                         // B = x[c][m], col m = l15
      bfr[0] = xb[(size_t)(kk + 0) * NTOK + mg];
      bfr[1] = xb[(size_t)(kk + 1) * NTOK + mg];
#pragma unroll
      for (int i = 0; i < 4; ++i) {
        const int o = og * 64 + i * 16 + l15;   // A = wq[o][c]
        v2f afr = *(const v2f*)(wq + (size_t)o * CDIM + kk);
        qa[i] = wmma4(afr, bfr, qa[i]);
      }
    }
    // transpose each 16x16 Q tile through LDS: D[o][m] -> A-frags of Q^T[m][c]
#pragma unroll
    for (int i = 0; i < 4; ++i) {
#pragma unroll
      for (int r = 0; r < 8; ++r) {
        const int oo = og * 64 + i * 16 + r + 8 * half;
        pbuf[wave][l15][r + 8 * half] = qa[i][r] + bq[oo];  // [m][o_local]
      }
      LDS_FENCE();
#pragma unroll
      for (int j = 0; j < 4; ++j)
        a_q[og * 16 + i * 4 + j] =
            *(const v2f*)&pbuf[wave][l15][4 * j + 2 * half];
      SCHED_FENCE();
    }
  }

  // ---- Main loop: stream 16-wide key tiles, online softmax, double-buffered ----
  v8f o_acc[16];                      // O[m][c] accumulators (rows m, cols c)
#pragma unroll
  for (int ct = 0; ct < 16; ++ct) o_acc[ct] = vzero8();
  float Mrow[8], Lrow[8];
#pragma unroll
  for (int r = 0; r < 8; ++r) { Mrow[r] = -3.0e38f; Lrow[r] = 0.f; }

  for (int ntile = 0; ntile < NT16; ++ntile) {
    const float* Kc = (ntile & 1) ? Kbuf1 : Kbuf0;
    const float* Vc = (ntile & 1) ? Vbuf1 : Vbuf0;
    if (ntile + 1 < NT16) {
      // prefetch next tile into the other buffer (its readers finished at the
      // barrier that ended iteration ntile-1)
      float* Kn = (ntile & 1) ? Kbuf0 : Kbuf1;
      float* Vn = (ntile & 1) ? Vbuf0 : Vbuf1;
      const int nb1 = (ntile + 1) * 16;
#pragma unroll 8
      for (int cc = crow; cc < CDIM; cc += 32) {
        async_copy_b128(Kn + cc * TILEW + n4, Kb + (size_t)cc * NTOK + nb1 + n4);
        async_copy_b128(Vn + cc * TILEW + n4, Vb + (size_t)cc * NTOK + nb1 + n4);
      }
      ASYNC_WAIT_PREFETCHED();   // current tile's 16 async ops are complete
    } else {
      ASYNC_WAIT_ALL();
    }
    __syncthreads();             // current tile visible to all waves

    // S = Q^T * K  (16m x 16n, k=256); 4 interleaved partial accumulators
    v8f sp[4];
    sp[0] = sp[1] = sp[2] = sp[3] = vzero8();
#pragma unroll
    for (int j = 0; j < 64; ++j) {
      const int kk = 4 * j + 2 * half;
      v2f bfr;                                   // B[k=c][n=l15]
      bfr[0] = Kc[(kk + 0) * TILEW + l15];
      bfr[1] = Kc[(kk + 1) * TILEW + l15];
      sp[j & 3] = wmma4(a_q[j], bfr, sp[j & 3]);
    }
    v8f s = (sp[0] + sp[1]) + (sp[2] + sp[3]);

    // online softmax; rows r+8*half private to each 16-lane group
    float alpha[8];
#pragma unroll
    for (int r = 0; r < 8; ++r) {
      float mx = s[r];
      mx = fmaxf(mx, __shfl_xor(mx, 1, 32));
      mx = fmaxf(mx, __shfl_xor(mx, 2, 32));
      mx = fmaxf(mx, __shfl_xor(mx, 4, 32));
      mx = fmaxf(mx, __shfl_xor(mx, 8, 32));
      const float Mn = fmaxf(Mrow[r], mx);
      alpha[r] = exp2f((Mrow[r] - Mn) * LOG2E);
      const float p = exp2f((s[r] - Mn) * LOG2E);
      float rs = p;
      rs += __shfl_xor(rs, 1, 32);
      rs += __shfl_xor(rs, 2, 32);
      rs += __shfl_xor(rs, 4, 32);
      rs += __shfl_xor(rs, 8, 32);
      Lrow[r] = Lrow[r] * alpha[r] + rs;
      Mrow[r] = Mn;
      s[r] = p;
    }
#pragma unroll
    for (int ct = 0; ct < 16; ++ct)
#pragma unroll
      for (int r = 0; r < 8; ++r) o_acc[ct][r] *= alpha[r];

    // relayout P: D-frag [m][n] -> A-frags via per-wave LDS
#pragma unroll
    for (int r = 0; r < 8; ++r)
      pbuf[wave][r + 8 * half][l15] = s[r];     // [m][n]
    LDS_FENCE();
    v2f a_p[4];
#pragma unroll
    for (int j = 0; j < 4; ++j)
      a_p[j] = *(const v2f*)&pbuf[wave][l15][4 * j + 2 * half];
    SCHED_FENCE();

    // O += P * V^T   (B[k=n][c] = Vt[c][n])
#pragma unroll
    for (int ct = 0; ct < 16; ++ct)
#pragma unroll
      for (int j = 0; j < 4; ++j) {
        v2f bfr = *(const v2f*)(Vc + (ct * 16 + l15) * TILEW + 4 * j + 2 * half);
        o_acc[ct] = wmma4(a_p[j], bfr, o_acc[ct]);
      }
    __syncthreads();   // all waves done with this tile before it is overwritten
  }

  // ---- Epilogue: Y = Wo * O^T + bo + x, per 16-c chunk; Wo staged in LDS ----
  float rinv[8];
#pragma unroll
  for (int r = 0; r < 8; ++r) rinv[r] = 1.0f / Lrow[r];

  float* wos = Kbuf0;                 // reuse 20KB of the staging area
  v8f y_acc[16];
#pragma unroll
  for (int ot = 0; ot < 16; ++ot) y_acc[ot] = vzero8();

#pragma unroll
  for (int cc = 0; cc < 16; ++cc) {
    __syncthreads();                  // previous chunk's readers are done
    // stage Wo[:, cc*16 .. cc*16+15] cooperatively (coalesced b128 loads)
#pragma unroll 8
    for (int oo = crow; oo < CDIM; oo += 32)
      *(float4*)(wos + oo * TILEW + n4) =
          *(const float4*)(wo + (size_t)oo * CDIM + cc * 16 + n4);
    // relayout this wave's O chunk: [m][c_local] through per-wave pbuf
#pragma unroll
    for (int r = 0; r < 8; ++r)
      pbuf[wave][r + 8 * half][l15] = o_acc[cc][r] * rinv[r];
    __syncthreads();                  // Wo chunk + (our) pbuf ready
    v2f b_o[4];
#pragma unroll
    for (int j = 0; j < 4; ++j)       // B[k=c][m=l15] = O[m][c]
      b_o[j] = *(const v2f*)&pbuf[wave][l15][4 * j + 2 * half];
    SCHED_FENCE();
#pragma unroll
    for (int ot = 0; ot < 16; ++ot)
#pragma unroll
      for (int j = 0; j < 4; ++j) {
        v2f afr = *(const v2f*)(wos + (ot * 16 + l15) * TILEW + 4 * j + 2 * half);
        y_acc[ot] = wmma4(afr, b_o[j], y_acc[ot]);
      }
  }

  float* yb = out + (size_t)b * CDIM * NTOK;
#pragma unroll
  for (int ot = 0; ot < 16; ++ot)
#pragma unroll
    for (int r = 0; r < 8; ++r) {
      const int o = ot * 16 + r + 8 * half;
      yb[(size_t)o * NTOK + mg] =
          y_acc[ot][r] + bo[o] + xb[(size_t)o * NTOK + mg];
    }
}

// ---------------------------------------------------------------------------
extern "C" void kernel_launch(void* const* d_in, const int* in_sizes, int n_in,
                              void* d_out, int out_size, void* d_ws, size_t ws_size,
                              hipStream_t stream) {
  (void)in_sizes; (void)n_in; (void)out_size; (void)ws_size;
  const float* x  = (const float*)d_in[0];
  const float* t  = (const float*)d_in[1];
  const float* wq = (const float*)d_in[2];
  const float* bq = (const float*)d_in[3];
  const float* wk = (const float*)d_in[4];
  const float* bk = (const float*)d_in[5];
  const float* wv = (const float*)d_in[6];
  const float* bv = (const float*)d_in[7];
  const float* wo = (const float*)d_in[8];
  const float* bo = (const float*)d_in[9];

  float* Kws = (float*)d_ws;                       // [B][C][N] f32, 8 MB
  float* Vws = Kws + (size_t)BSZ * CDIM * NTOK;    // [B][C][N] f32, 8 MB

  dim3 g1(NTOK / 64, BSZ, 2);
  kv_proj_kernel<<<g1, dim3(128, 1, 1), 0, stream>>>(t, wk, bk, wv, bv, Kws, Vws);

  const size_t smem = (size_t)4 * CDIM * TILEW * sizeof(float);  // 80 KB
  dim3 g2(NTOK / 64, BSZ, 1);
  attn_kernel<<<g2, dim3(128, 1, 1), smem, stream>>>(x, wq, bq, wo, bo,
                                                     Kws, Vws, (float*)d_out);
}